// RAA_1606317769475
// MI455X (gfx1250) — compile-verified
//
#include <hip/hip_runtime.h>
#include <hip/hip_bf16.h>
#include <math.h>

#define NN 4096
#define KK 25
#define TSTRIDE 32   // temp row stride (25 valid + zero padding -> k loop to 28 is safe)

typedef float v2f __attribute__((ext_vector_type(2)));
typedef float v8f __attribute__((ext_vector_type(8)));

// ---------------- init: zero output scalar and M scratch ----------------
__global__ void init_kernel(float* __restrict__ out, float* __restrict__ M) {
    int t = blockIdx.x * blockDim.x + threadIdx.x;
    if (t == 0) out[0] = 0.0f;
    if (t < KK * KK) M[t] = 0.0f;
}

// ---------------- softmax over rows of Z (K rows of length N) ----------------
__global__ void softmax_z_kernel(const float* __restrict__ Z, float* __restrict__ Zs) {
    const int row = blockIdx.x;                 // 0..24
    const float* zr = Z + (size_t)row * NN;
    float* outr = Zs + (size_t)row * NN;
    __shared__ float red[256];

    float m = -3.4e38f;
    for (int idx = threadIdx.x; idx < NN; idx += 256) m = fmaxf(m, zr[idx]);
    red[threadIdx.x] = m; __syncthreads();
    for (int s = 128; s > 0; s >>= 1) {
        if (threadIdx.x < s) red[threadIdx.x] = fmaxf(red[threadIdx.x], red[threadIdx.x + s]);
        __syncthreads();
    }
    m = red[0]; __syncthreads();

    float sum = 0.0f;
    for (int idx = threadIdx.x; idx < NN; idx += 256) sum += __expf(zr[idx] - m);
    red[threadIdx.x] = sum; __syncthreads();
    for (int s = 128; s > 0; s >>= 1) {
        if (threadIdx.x < s) red[threadIdx.x] += red[threadIdx.x + s];
        __syncthreads();
    }
    const float inv = 1.0f / red[0];

    for (int idx = threadIdx.x; idx < NN; idx += 256)
        outr[idx] = __expf(zr[idx] - m) * inv;
}

// ---------------- softmax over rows of C (N rows of length K) ----------------
__global__ void softmax_c_kernel(const float* __restrict__ C, float* __restrict__ Cs) {
    int n = blockIdx.x * blockDim.x + threadIdx.x;
    if (n >= NN) return;
    const float* cr = C + (size_t)n * KK;
    float v[KK];
    float m = -3.4e38f;
    #pragma unroll
    for (int k = 0; k < KK; ++k) { v[k] = cr[k]; m = fmaxf(m, v[k]); }
    float s = 0.0f;
    #pragma unroll
    for (int k = 0; k < KK; ++k) { v[k] = __expf(v[k] - m); s += v[k]; }
    const float inv = 1.0f / s;
    float* o = Cs + (size_t)n * KK;
    #pragma unroll
    for (int k = 0; k < KK; ++k) o[k] = v[k] * inv;
}

// ---------------- M = Zs @ Cs  (K x K), chunked over n with atomics ----------------
__global__ void gemm_m_kernel(const float* __restrict__ Zs, const float* __restrict__ Cs,
                              float* __restrict__ M) {
    int t = threadIdx.x;
    if (t >= KK * KK) return;
    const int p = t / KK, q = t % KK;
    const int n0 = blockIdx.x * 64;
    float acc = 0.0f;
    for (int n = n0; n < n0 + 64; ++n)
        acc += Zs[(size_t)p * NN + n] * Cs[(size_t)n * KK + q];
    atomicAdd(&M[t], acc);
}

// ---------------- temp[n][k] = sum_p M[k][p] * Zs[p][n]; plus ||t_n||^2 and sum_k t_n ---
__global__ void temp_kernel(const float* __restrict__ Zs, const float* __restrict__ M,
                            float* __restrict__ temp, float* __restrict__ sqn,
                            float* __restrict__ ss) {
    __shared__ float Msh[KK * KK];
    for (int t = threadIdx.x; t < KK * KK; t += blockDim.x) Msh[t] = M[t];
    __syncthreads();
    int n = blockIdx.x * blockDim.x + threadIdx.x;
    if (n >= NN) return;
    float z[KK];
    #pragma unroll
    for (int p = 0; p < KK; ++p) z[p] = Zs[(size_t)p * NN + n];   // coalesced per p
    float sq = 0.0f, s = 0.0f;
    float* tr = temp + (size_t)n * TSTRIDE;
    #pragma unroll
    for (int k = 0; k < KK; ++k) {
        float acc = 0.0f;
        #pragma unroll
        for (int p = 0; p < KK; ++p) acc += Msh[k * KK + p] * z[p];
        tr[k] = acc; sq += acc * acc; s += acc;
    }
    #pragma unroll
    for (int k = KK; k < TSTRIDE; ++k) tr[k] = 0.0f;   // zero pad -> WMMA K=28 safe
    sqn[n] = sq; ss[n] = s;
}

// ---------------- main pairwise kernel: WMMA Gram tile + epilogue ----------------
// grid: 8192 blocks x 256 threads (8 waves); one 16x16 tile of the N x N pair
// matrix per wave; 256x256 tiles total. All lanes active (EXEC==all-1s for WMMA).
__global__ void __launch_bounds__(256)
pair_ll_kernel(const float* __restrict__ A, const float* __restrict__ beta,
               const float* __restrict__ a_scalar, const float* __restrict__ temp,
               const float* __restrict__ sqn, const float* __restrict__ ss,
               float* __restrict__ out) {
    const int lane = threadIdx.x & 31;
    const int wave = threadIdx.x >> 5;
    const int gw = blockIdx.x * 8 + wave;        // global wave/tile id, 0..65535
    const int i_base = (gw >> 8) * 16;
    const int j_base = (gw & 255) * 16;
    const int half = lane >> 4;                  // 0: lanes 0-15, 1: lanes 16-31
    const int l16 = lane & 15;

    const float a0 = a_scalar[0];

    // Gram tile G = T_i (16xK) * T_j^T (Kx16) via 7 steps of f32 WMMA 16x16x4.
    // A-layout: lane<16 row M=l16 holds K={ks,ks+1}; lane>=16 holds K={ks+2,ks+3}.
    // B-layout mirrors with column N=l16: B[k][n] = temp[j_base+n][k].
    const float* ta = temp + (size_t)(i_base + l16) * TSTRIDE + 2 * half;
    const float* tb = temp + (size_t)(j_base + l16) * TSTRIDE + 2 * half;

    v8f c = {0.f, 0.f, 0.f, 0.f, 0.f, 0.f, 0.f, 0.f};
    #pragma unroll
    for (int ks = 0; ks < 28; ks += 4) {
        v2f av, bv;
        av.x = ta[ks]; av.y = ta[ks + 1];
        bv.x = tb[ks]; bv.y = tb[ks + 1];
        c = __builtin_amdgcn_wmma_f32_16x16x4_f32(false, av, false, bv,
                                                  (short)0, c, false, false);
    }

    // Epilogue: ||ti-tj+e||^2 = sqn_i + sqn_j - 2G + 2e(ss_i-ss_j) + K e^2
    const int j = j_base + l16;
    const float sqn_j = sqn[j];
    const float ss_j  = ss[j];
    const float beta_j = beta[j];
    const float e = 1e-6f;
    const float ke2 = (float)KK * e * e;

    float acc = 0.0f;
    #pragma unroll
    for (int r = 0; r < 8; ++r) {
        const int i = i_base + r + 8 * half;     // C/D layout: M = r + 8*half, N = l16
        const float g = c[r];
        float d2 = sqn[i] + sqn_j - 2.0f * g + 2.0f * e * (ss[i] - ss_j) + ke2;
        d2 = fmaxf(d2, 0.0f);
        const float z = sqrtf(d2);
        const float theta = beta_j - a0 * z;
        const float Aij = A[(size_t)i * NN + j]; // coalesced 64B per half-wave
        const float sp = fmaxf(theta, 0.0f) + log1pf(__expf(-fabsf(theta)));
        acc += theta * Aij - sp;
    }

    // block reduction -> single atomic per block
    __shared__ float red[256];
    red[threadIdx.x] = acc;
    __syncthreads();
    for (int s = 128; s > 0; s >>= 1) {
        if (threadIdx.x < s) red[threadIdx.x] += red[threadIdx.x + s];
        __syncthreads();
    }
    if (threadIdx.x == 0) atomicAdd(out, red[0]);
}

extern "C" void kernel_launch(void* const* d_in, const int* in_sizes, int n_in,
                              void* d_out, int out_size, void* d_ws, size_t ws_size,
                              hipStream_t stream) {
    const float* A    = (const float*)d_in[0];   // (N,N)
    const float* beta = (const float*)d_in[1];   // (N,)
    const float* a    = (const float*)d_in[2];   // (1,)
    const float* Z    = (const float*)d_in[3];   // (K,N)
    const float* C    = (const float*)d_in[4];   // (N,K)
    float* out = (float*)d_out;                  // scalar

    float* ws   = (float*)d_ws;
    float* Zs   = ws;                            // 25*4096
    float* Cs   = Zs + KK * NN;                  // 4096*25
    float* M    = Cs + NN * KK;                  // 640 (25*25 padded)
    float* temp = M + 640;                       // 4096*32
    float* sqn  = temp + (size_t)NN * TSTRIDE;   // 4096
    float* ss   = sqn + NN;                      // 4096

    init_kernel<<<1, 640, 0, stream>>>(out, M);
    softmax_z_kernel<<<KK, 256, 0, stream>>>(Z, Zs);
    softmax_c_kernel<<<NN / 256, 256, 0, stream>>>(C, Cs);
    gemm_m_kernel<<<NN / 64, 640, 0, stream>>>(Zs, Cs, M);
    temp_kernel<<<NN / 256, 256, 0, stream>>>(Zs, M, temp, sqn, ss);
    pair_ll_kernel<<<(NN / 16) * (NN / 16) / 8, 256, 0, stream>>>(
        A, beta, a, temp, sqn, ss, out);
}